// SoftGouraudShader_16406775070837
// MI455X (gfx1250) — compile-verified
//
#include <hip/hip_runtime.h>
#include <math.h>

// ---------------------------------------------------------------------------
// Problem constants (from the reference)
// ---------------------------------------------------------------------------
namespace {
constexpr int   kV        = 100000;
constexpr int   kF        = 200000;
constexpr int   kNPix     = 4 * 512 * 512;   // N*H*W
constexpr float kSigmaInv = 1.0e4f;          // 1/SIGMA
constexpr float kGammaInv = 1.0e4f;          // 1/GAMMA
constexpr float kZFar     = 100.0f;
constexpr float kZRngInv  = 1.0f / 99.0f;    // 1/(ZFAR-ZNEAR)
constexpr float kEps      = 1e-10f;
}

// ---------------------------------------------------------------------------
// gfx1250 async global->LDS path (guarded; falls back to direct b128 loads)
// Builtin signature (from hipcc diagnostic): (v4i AS1*, v4i AS3*, Iimm, Iimm)
// ---------------------------------------------------------------------------
#if defined(__gfx1250__) && __has_builtin(__builtin_amdgcn_global_load_async_to_lds_b128) && __has_builtin(__builtin_amdgcn_s_wait_asynccnt)
#define USE_ASYNC_LDS 1
#else
#define USE_ASYNC_LDS 0
#endif

#if USE_ASYNC_LDS
typedef __attribute__((ext_vector_type(4))) int v4i;

__device__ __forceinline__ void async_copy_b128(void* lds_dst, const void* gsrc) {
  __builtin_amdgcn_global_load_async_to_lds_b128(
      (__attribute__((address_space(1))) v4i*)const_cast<void*>(gsrc),
      (__attribute__((address_space(3))) v4i*)lds_dst,
      /*offset=*/0, /*cpol=*/0);
}
#endif

// ---------------------------------------------------------------------------
// Kernel A: per-vertex Phong (gouraud) shading -> float4 table in ws
// ---------------------------------------------------------------------------
__global__ __launch_bounds__(256) void shade_vertices(
    const float* __restrict__ verts, const float* __restrict__ normals,
    const float* __restrict__ vcolors,
    const float* __restrict__ light_pos, const float* __restrict__ cam_pos,
    const float* __restrict__ ambient, const float* __restrict__ diffuse,
    const float* __restrict__ specular,
    float4* __restrict__ vcol_out)
{
  const int i = blockIdx.x * 256 + threadIdx.x;
  if (i >= kV) return;

  const float px = verts[3*i+0], py = verts[3*i+1], pz = verts[3*i+2];
  float nx = normals[3*i+0], ny = normals[3*i+1], nz = normals[3*i+2];
  const float nn = fmaxf(sqrtf(nx*nx + ny*ny + nz*nz), 1e-6f);
  nx /= nn; ny /= nn; nz /= nn;

  float lx = light_pos[0]-px, ly = light_pos[1]-py, lz = light_pos[2]-pz;
  const float ln = fmaxf(sqrtf(lx*lx + ly*ly + lz*lz), 1e-6f);
  lx /= ln; ly /= ln; lz /= ln;

  float vx = cam_pos[0]-px, vy = cam_pos[1]-py, vz = cam_pos[2]-pz;
  const float vn = fmaxf(sqrtf(vx*vx + vy*vy + vz*vz), 1e-6f);
  vx /= vn; vy /= vn; vz /= vn;

  const float ndl   = nx*lx + ny*ly + nz*lz;
  const float cos_l = fmaxf(ndl, 0.0f);
  const float rx = 2.0f*ndl*nx - lx;
  const float ry = 2.0f*ndl*ny - ly;
  const float rz = 2.0f*ndl*nz - lz;
  const float cos_s = fmaxf(rx*vx + ry*vy + rz*vz, 0.0f);
  // cos_s^64 via 6 squarings
  float s = cos_s * cos_s;                  // ^2
  s *= s; s *= s; s *= s; s *= s; s *= s;   // ^64

  float4 o;
  o.x = vcolors[3*i+0]*(ambient[0] + diffuse[0]*cos_l) + specular[0]*s;
  o.y = vcolors[3*i+1]*(ambient[1] + diffuse[1]*cos_l) + specular[1]*s;
  o.z = vcolors[3*i+2]*(ambient[2] + diffuse[2]*cos_l) + specular[2]*s;
  o.w = 0.0f;
  vcol_out[i] = o;
}

// ---------------------------------------------------------------------------
// Kernel B: gather shaded colors per face -> 48B rows (3 x float4) in ws.
// This table (9.6 MB) fits in the 192 MB L2, turning the per-fragment gather
// into a single L2-resident b128x3 load instead of a dependent chain.
// ---------------------------------------------------------------------------
__global__ __launch_bounds__(256) void build_face_colors(
    const int* __restrict__ faces, const float4* __restrict__ vcol,
    float4* __restrict__ fcol)
{
  const int f = blockIdx.x * 256 + threadIdx.x;
  if (f >= kF) return;
  const int a = faces[3*f+0], b = faces[3*f+1], c = faces[3*f+2];
  fcol[3*f+0] = vcol[a];
  fcol[3*f+1] = vcol[b];
  fcol[3*f+2] = vcol[c];
}

// ---------------------------------------------------------------------------
// Kernel C: per-pixel softmax RGB blend. One thread per pixel, K=8 unrolled.
// Streamed inputs staged per-block via gfx1250 async global->LDS b128 copies
// (chunk-transposed layout -> bank-conflict-free ds_load_b128 reads), with a
// direct coalesced-b128 fallback when the async builtins are unavailable.
// ---------------------------------------------------------------------------
__global__ __launch_bounds__(256) void blend_pixels(
    const int*   __restrict__ p2f_g,  const float* __restrict__ bary_g,
    const float* __restrict__ dist_g, const float* __restrict__ zbuf_g,
    const float4* __restrict__ fcol,  const int*   __restrict__ faces,
    const float4* __restrict__ vcol,  int use_fcol,
    float4* __restrict__ out)
{
  const int t = threadIdx.x;
  const int P = blockIdx.x * 256 + t;
  if (P >= kNPix) return;   // exact division; never taken

  // Per-pixel contiguous byte regions (all 16B aligned).
  const char* gp = (const char*)p2f_g  + (size_t)P * 32;   // 8 ints
  const char* gb = (const char*)bary_g + (size_t)P * 96;   // 24 floats
  const char* gd = (const char*)dist_g + (size_t)P * 32;   // 8 floats
  const char* gz = (const char*)zbuf_g + (size_t)P * 32;   // 8 floats

  int4   pc[2];
  float4 bc[6];
  float4 dc[2];
  float4 zc[2];

#if USE_ASYNC_LDS
  __shared__ __align__(16) unsigned char smem[48 * 1024];
  char* s0 = (char*)smem;                         // p2f  : [2][256] x 16B
  char* s1 = (char*)smem + 8192;                  // bary : [6][256] x 16B
  char* s2 = (char*)smem + 8192 + 24576;          // dist : [2][256] x 16B
  char* s3 = (char*)smem + 8192 + 24576 + 8192;   // zbuf : [2][256] x 16B
  #pragma unroll
  for (int j = 0; j < 2; ++j) async_copy_b128(s0 + (size_t)(j*256 + t)*16, gp + j*16);
  #pragma unroll
  for (int j = 0; j < 6; ++j) async_copy_b128(s1 + (size_t)(j*256 + t)*16, gb + j*16);
  #pragma unroll
  for (int j = 0; j < 2; ++j) async_copy_b128(s2 + (size_t)(j*256 + t)*16, gd + j*16);
  #pragma unroll
  for (int j = 0; j < 2; ++j) async_copy_b128(s3 + (size_t)(j*256 + t)*16, gz + j*16);
  __builtin_amdgcn_s_wait_asynccnt(0);
  #pragma unroll
  for (int j = 0; j < 2; ++j) pc[j] = ((const int4*)  s0)[j*256 + t];
  #pragma unroll
  for (int j = 0; j < 6; ++j) bc[j] = ((const float4*)s1)[j*256 + t];
  #pragma unroll
  for (int j = 0; j < 2; ++j) dc[j] = ((const float4*)s2)[j*256 + t];
  #pragma unroll
  for (int j = 0; j < 2; ++j) zc[j] = ((const float4*)s3)[j*256 + t];
#else
  #pragma unroll
  for (int j = 0; j < 2; ++j) pc[j] = *(const int4*)  (gp + j*16);
  #pragma unroll
  for (int j = 0; j < 6; ++j) bc[j] = *(const float4*)(gb + j*16);
  #pragma unroll
  for (int j = 0; j < 2; ++j) dc[j] = *(const float4*)(gd + j*16);
  #pragma unroll
  for (int j = 0; j < 2; ++j) zc[j] = *(const float4*)(gz + j*16);
#endif

  int   pf[8] = { pc[0].x, pc[0].y, pc[0].z, pc[0].w,
                  pc[1].x, pc[1].y, pc[1].z, pc[1].w };
  float df[8] = { dc[0].x, dc[0].y, dc[0].z, dc[0].w,
                  dc[1].x, dc[1].y, dc[1].z, dc[1].w };
  float zf[8] = { zc[0].x, zc[0].y, zc[0].z, zc[0].w,
                  zc[1].x, zc[1].y, zc[1].z, zc[1].w };
  float bf[24];
  #pragma unroll
  for (int j = 0; j < 6; ++j) {
    bf[4*j+0] = bc[j].x; bf[4*j+1] = bc[j].y;
    bf[4*j+2] = bc[j].z; bf[4*j+3] = bc[j].w;
  }

  // Issue L2 prefetches for the random gather targets early.
  if (use_fcol) {
    #pragma unroll
    for (int k = 0; k < 8; ++k)
      if (pf[k] >= 0) __builtin_prefetch((const void*)(fcol + 3*(size_t)pf[k]), 0, 0);
  } else {
    #pragma unroll
    for (int k = 0; k < 8; ++k)
      if (pf[k] >= 0) __builtin_prefetch((const void*)(faces + 3*(size_t)pf[k]), 0, 0);
  }

  // prob = sigmoid(-d/sigma)*mask ; z_inv = (zfar-z)/(zfar-znear)*mask
  float prob[8], zin[8];
  float one_minus = 1.0f;
  float zmax = kEps;
  #pragma unroll
  for (int k = 0; k < 8; ++k) {
    const bool m = (pf[k] >= 0);
    float p = 1.0f / (1.0f + expf(df[k] * kSigmaInv));  // sigmoid(-d/sigma)
    p = m ? p : 0.0f;
    prob[k] = p;
    one_minus *= (1.0f - p);
    const float zi = m ? (kZFar - zf[k]) * kZRngInv : 0.0f;
    zin[k] = zi;
    zmax = fmaxf(zmax, zi);
  }
  const float alpha = 1.0f - one_minus;
  const float delta = expf((kEps - zmax) * kGammaInv);

  float denom = delta;
  float r = 0.0f, g = 0.0f, b = 0.0f;
  #pragma unroll
  for (int k = 0; k < 8; ++k) {
    const float w = prob[k] * expf((zin[k] - zmax) * kGammaInv);
    denom += w;
    if (pf[k] >= 0) {
      float4 c0, c1, c2;
      if (use_fcol) {
        const float4* row = fcol + 3*(size_t)pf[k];
        c0 = row[0]; c1 = row[1]; c2 = row[2];
      } else {
        const int f  = pf[k];
        const int ia = faces[3*f+0], ib = faces[3*f+1], ic = faces[3*f+2];
        c0 = vcol[ia]; c1 = vcol[ib]; c2 = vcol[ic];
      }
      const float b0 = bf[3*k+0], b1 = bf[3*k+1], b2 = bf[3*k+2];
      r += w * (b0*c0.x + b1*c1.x + b2*c2.x);
      g += w * (b0*c0.y + b1*c1.y + b2*c2.y);
      b += w * (b0*c0.z + b1*c1.z + b2*c2.z);
    }
  }

  const float inv = 1.0f / denom;
  float4 o;
  o.x = (r + delta) * inv;   // background = (1,1,1)
  o.y = (g + delta) * inv;
  o.z = (b + delta) * inv;
  o.w = alpha;
  out[P] = o;
}

// ---------------------------------------------------------------------------
// Host launcher
// ---------------------------------------------------------------------------
extern "C" void kernel_launch(void* const* d_in, const int* in_sizes, int n_in,
                              void* d_out, int out_size, void* d_ws, size_t ws_size,
                              hipStream_t stream) {
  (void)in_sizes; (void)n_in; (void)out_size;

  const float* verts       = (const float*)d_in[0];
  const float* normals     = (const float*)d_in[1];
  const float* vcolors     = (const float*)d_in[2];
  const int*   faces       = (const int*)  d_in[3];
  const int*   p2f         = (const int*)  d_in[4];
  const float* bary        = (const float*)d_in[5];
  const float* dists       = (const float*)d_in[6];
  const float* zbuf        = (const float*)d_in[7];
  const float* light_pos   = (const float*)d_in[8];
  const float* cam_pos     = (const float*)d_in[9];
  const float* ambient     = (const float*)d_in[10];
  const float* diffuse     = (const float*)d_in[11];
  const float* specular    = (const float*)d_in[12];

  const size_t vcol_bytes = (size_t)kV * sizeof(float4);        // 1.6 MB
  const size_t fcol_bytes = (size_t)kF * 3 * sizeof(float4);    // 9.6 MB

  float4* ws_vcol = (float4*)d_ws;
  float4* ws_fcol = (float4*)((char*)d_ws + vcol_bytes);        // 256B-aligned offset

  const int use_fcol = (ws_size >= vcol_bytes + fcol_bytes) ? 1 : 0;

  shade_vertices<<<(kV + 255) / 256, 256, 0, stream>>>(
      verts, normals, vcolors, light_pos, cam_pos, ambient, diffuse, specular,
      ws_vcol);

  if (use_fcol) {
    build_face_colors<<<(kF + 255) / 256, 256, 0, stream>>>(
        faces, ws_vcol, ws_fcol);
  }

  blend_pixels<<<kNPix / 256, 256, 0, stream>>>(
      p2f, bary, dists, zbuf, ws_fcol, faces, ws_vcol, use_fcol,
      (float4*)d_out);
}